// DinkNet_dgl_19026705121766
// MI455X (gfx1250) — compile-verified
//
#include <hip/hip_runtime.h>
#include <hip/hip_bf16.h>

typedef __attribute__((ext_vector_type(2))) float v2f;
typedef __attribute__((ext_vector_type(8))) float v8f;
typedef __attribute__((ext_vector_type(4))) unsigned int v4u;
typedef __attribute__((ext_vector_type(8))) int v8i;
typedef __attribute__((ext_vector_type(4))) int v4i;

#define N_IN 512
#define N_H  128
#define KC   64    // K-chunk staged in LDS
#define LDW  132   // LDS row stride (floats): 128 data + 4 pad (= TDM pad_amount)

#if __has_builtin(__builtin_amdgcn_tensor_load_to_lds)
#define HAVE_TDM 1
#else
#define HAVE_TDM 0
#endif

// ---------------------------------------------------------------- utilities
__global__ __launch_bounds__(256) void zero_k(float* __restrict__ p, size_t n4) {
    size_t i = (size_t)blockIdx.x * blockDim.x + threadIdx.x;
    size_t stride = (size_t)gridDim.x * blockDim.x;
    float4 z = make_float4(0.f, 0.f, 0.f, 0.f);
    for (; i < n4; i += stride) reinterpret_cast<float4*>(p)[i] = z;
}

// degree counting: float atomics into pre-zeroed cnt arrays
__global__ __launch_bounds__(256) void count_deg(const int* __restrict__ src,
                                                 const int* __restrict__ dst,
                                                 float* __restrict__ cnt_out,
                                                 float* __restrict__ cnt_in, int e) {
    int i = blockIdx.x * 256 + threadIdx.x;
    if (i < e) {
        atomicAdd(cnt_out + src[i], 1.0f);
        atomicAdd(cnt_in  + dst[i], 1.0f);
    }
}

// cnt -> clip(cnt,1)^-0.5 in place, over both arrays (2n contiguous floats)
__global__ __launch_bounds__(256) void norm_k(float* __restrict__ buf, int n2) {
    int i = blockIdx.x * 256 + threadIdx.x;
    if (i < n2) {
        float c = buf[i];
        c = c < 1.0f ? 1.0f : c;
        buf[i] = rsqrtf(c);
    }
}

// wsum[k] = sum_j mlp_W[k][j]; wsum[128] = sum(mlp_b)
__global__ __launch_bounds__(256) void wsum_k(const float* __restrict__ mlp_W,
                                              const float* __restrict__ mlp_b,
                                              float* __restrict__ wsum) {
    int r = threadIdx.x;
    if (r < N_H) {
        float s = 0.f;
        #pragma unroll 8
        for (int j = 0; j < N_H; ++j) s += mlp_W[r * N_H + j];
        wsum[r] = s;
    } else if (r == N_H) {
        float s = 0.f;
        for (int j = 0; j < N_H; ++j) s += mlp_b[j];
        wsum[N_H] = s;
    }
}

// ---------------------------------------------------------------- GEMM h = x @ W
// 256 threads = 8 waves; each wave owns 16 rows x 128 cols (8 WMMA C-tiles).
// fp32 WMMA 16x16x4 lane layout (ISA 7.12.2): lanes 0-15 hold M=lane with
// VGPR0=K0,VGPR1=K1; lanes 16-31 hold M=lane-16 with K2,K3. B mirrored over N.
// W chunks are staged into LDS by the Tensor Data Mover (when available),
// using D# padding (128 DWORDs interval, 4 DWORD pad) to produce LDW=132.
__global__ __launch_bounds__(256) void gemm_h(const float* __restrict__ x,
                                              const float* __restrict__ W,
                                              float* __restrict__ h, int n) {
    __shared__ float Wt[KC * LDW];

    const int lane    = threadIdx.x & 31;
    const int wave    = threadIdx.x >> 5;
    const int halfSel = lane >> 4;      // 0: K0/K1 half, 1: K2/K3 half
    const int ln      = lane & 15;
    const int rowBase = blockIdx.x * 128 + wave * 16;

    v8f acc[8];
    for (int t = 0; t < 8; ++t)
        for (int j = 0; j < 8; ++j) acc[t][j] = 0.f;

    int arow = rowBase + ln;
    if (arow >= n) arow = n - 1;        // clamp; tail results are never stored
    const float* xrow = x + (size_t)arow * N_IN;

    for (int kc = 0; kc < N_IN; kc += KC) {
        __syncthreads();                // previous chunk fully consumed

#if HAVE_TDM
        if (wave == 0) {
            // ---- Tensor DMA: W[kc..kc+KC) x 128 -> LDS (padded rows) ----
            unsigned long long ga = (unsigned long long)(size_t)(W + (size_t)kc * N_H);
            v4u g0;
            g0[0] = 1u;                                   // count=1, user mode
            g0[1] = (unsigned int)(size_t)(void*)Wt;      // lds_addr (LDS offset)
            g0[2] = (unsigned int)ga;                     // global_addr[31:0]
            g0[3] = (unsigned int)(ga >> 32) | (2u << 30);// global_addr[56:32] | type=2
            v8i g1 = (v8i)0;
            // data_size=4B(2), pad_enable, pad_interval=6 (128 DW), pad_amount=3 (4 DW)
            g1[0] = (int)((2u << 16) | (1u << 20) | (6u << 22) | (3u << 25));
            g1[1] = (int)((unsigned)N_H << 16);           // tensor_dim0 = 128
            g1[2] = (int)((unsigned)N_IN << 16);          // tensor_dim1 = 512
            g1[3] = (int)((unsigned)N_H << 16);           // tile_dim0 = 128
            g1[4] = KC;                                   // tile_dim1 = 64
            g1[5] = N_H;                                  // tensor_dim0_stride = 128
            v4i gz = (v4i)0;
#if defined(__clang_major__) && __clang_major__ >= 23
            __builtin_amdgcn_tensor_load_to_lds(g0, g1, gz, gz, (v8i)0, 0);
#else
            __builtin_amdgcn_tensor_load_to_lds(g0, g1, gz, gz, 0);
#endif
            __builtin_amdgcn_s_wait_tensorcnt(0);
        }
#else
        // ---- fallback: cooperative vector load of the W chunk ----
        for (int i = threadIdx.x; i < (KC * N_H) / 4; i += 256) {
            int kk = i >> 5;            // (i*4)/128
            int nn = (i * 4) & (N_H - 1);
            float4 wv = *reinterpret_cast<const float4*>(W + (size_t)(kc + kk) * N_H + nn);
            float* d = Wt + kk * LDW + nn;
            d[0] = wv.x; d[1] = wv.y; d[2] = wv.z; d[3] = wv.w;
        }
#endif
        __syncthreads();

        if (kc + KC < N_IN)
            __builtin_prefetch(xrow + kc + KC, 0, 1);     // global_prefetch_b8

        #pragma unroll 4
        for (int kk = 0; kk < KC; kk += 4) {
            const int kq = kk + halfSel * 2;
            v2f a;
            a.x = xrow[kc + kq];
            a.y = xrow[kc + kq + 1];
            #pragma unroll
            for (int t = 0; t < 8; ++t) {
                v2f b;
                b.x = Wt[kq * LDW + t * 16 + ln];
                b.y = Wt[(kq + 1) * LDW + t * 16 + ln];
                acc[t] = __builtin_amdgcn_wmma_f32_16x16x4_f32(
                    /*neg_a=*/false, a, /*neg_b=*/false, b,
                    /*c_mod=*/(short)0, acc[t],
                    /*reuse_a=*/false, /*reuse_b=*/false);
            }
        }
    }

    // C/D layout: VGPR r -> row rowBase+r (lanes 0-15), rowBase+8+r (lanes 16-31)
    const int srowBase = rowBase + halfSel * 8;
    #pragma unroll
    for (int r = 0; r < 8; ++r) {
        int row = srowBase + r;
        if (row < n) {
            float* hp = h + (size_t)row * N_H + ln;
            #pragma unroll
            for (int t = 0; t < 8; ++t) hp[t * 16] = acc[t][r];
        }
    }
}

// ---------------------------------------------------------------- SpMM scatter
// one wave32 per edge; lane handles 4 features (float4). agg[dst] += c * h[row]
template <bool PERM>
__global__ __launch_bounds__(256) void spmm(const float* __restrict__ h,
                                            const int* __restrict__ src,
                                            const int* __restrict__ dst,
                                            const int* __restrict__ perm,
                                            const float* __restrict__ norm_out,
                                            float* __restrict__ agg, int e) {
    int edge = blockIdx.x * 8 + (threadIdx.x >> 5);
    if (edge >= e) return;
    int lane = threadIdx.x & 31;
    int s = src[edge];
    int d = dst[edge];
    float c = norm_out[s];
    int row = PERM ? perm[s] : s;
    float4 v = *reinterpret_cast<const float4*>(h + (size_t)row * N_H + lane * 4);
    float* ap = agg + (size_t)d * N_H + lane * 4;
    atomicAdd(ap + 0, c * v.x);
    atomicAdd(ap + 1, c * v.y);
    atomicAdd(ap + 2, c * v.z);
    atomicAdd(ap + 3, c * v.w);
}

// ---------------------------------------------------------------- finalize
// z = prelu(agg*norm_in + b); out[node] = dot(z, wsum) + wsum[128]
__global__ __launch_bounds__(256) void finalize_k(const float* __restrict__ agg,
                                                  const float* __restrict__ norm_in,
                                                  const float* __restrict__ b,
                                                  const float* __restrict__ alpha,
                                                  const float* __restrict__ wsum,
                                                  float* __restrict__ out, int n) {
    int node = blockIdx.x * 8 + (threadIdx.x >> 5);
    if (node >= n) return;
    int lane = threadIdx.x & 31;
    float ni = norm_in[node];
    float4 a  = *reinterpret_cast<const float4*>(agg + (size_t)node * N_H + lane * 4);
    float4 bb = *reinterpret_cast<const float4*>(b + lane * 4);
    float4 al = *reinterpret_cast<const float4*>(alpha + lane * 4);
    float4 wv = *reinterpret_cast<const float4*>(wsum + lane * 4);

    float z0 = fmaf(a.x, ni, bb.x); z0 = z0 >= 0.f ? z0 : al.x * z0;
    float z1 = fmaf(a.y, ni, bb.y); z1 = z1 >= 0.f ? z1 : al.y * z1;
    float z2 = fmaf(a.z, ni, bb.z); z2 = z2 >= 0.f ? z2 : al.z * z2;
    float z3 = fmaf(a.w, ni, bb.w); z3 = z3 >= 0.f ? z3 : al.w * z3;

    float s = z0 * wv.x + z1 * wv.y + z2 * wv.z + z3 * wv.w;
    for (int off = 16; off > 0; off >>= 1) s += __shfl_down(s, off, 32);
    if (lane == 0) out[node] = s + wsum[N_H];
}

// ---------------------------------------------------------------- launcher
extern "C" void kernel_launch(void* const* d_in, const int* in_sizes, int n_in,
                              void* d_out, int out_size, void* d_ws, size_t ws_size,
                              hipStream_t stream) {
    const float* x     = (const float*)d_in[0];
    const int*   src   = (const int*)  d_in[1];
    const int*   dst   = (const int*)  d_in[2];
    const int*   perm  = (const int*)  d_in[3];
    const float* W     = (const float*)d_in[4];
    const float* b     = (const float*)d_in[5];
    const float* alpha = (const float*)d_in[6];
    const float* mlpW  = (const float*)d_in[7];
    const float* mlpb  = (const float*)d_in[8];

    const int n = in_sizes[3];   // perm has N elements
    const int e = in_sizes[1];   // src has E elements
    float* out = (float*)d_out;

    // workspace layout
    float* ws       = (float*)d_ws;
    float* h        = ws;                                  // n*128
    float* agg      = h + (size_t)n * N_H;                 // n*128 (reused)
    float* norm_out = agg + (size_t)n * N_H;               // n
    float* norm_in  = norm_out + n;                        // n
    float* wsum     = norm_in + n;                         // 129

    const int ZB = 2048;

    // 1) degree norms
    size_t nrm4 = ((size_t)2 * n + 3) / 4;                 // zero both norm arrays
    int ngrid = (int)((nrm4 + 255) / 256); if (ngrid > ZB) ngrid = ZB;
    zero_k<<<ngrid, 256, 0, stream>>>(norm_out, nrm4);
    count_deg<<<(e + 255) / 256, 256, 0, stream>>>(src, dst, norm_out, norm_in, e);
    norm_k<<<(2 * n + 255) / 256, 256, 0, stream>>>(norm_out, 2 * n);

    // 2) collapsed MLP vector
    wsum_k<<<1, 256, 0, stream>>>(mlpW, mlpb, wsum);

    // 3) h = x @ W  (fp32 WMMA, TDM-staged W tiles)
    gemm_h<<<(n + 127) / 128, 256, 0, stream>>>(x, W, h, n);

    // 4) encoder 1: agg = A_norm @ h ; logit[0:n]
    size_t agg4 = (size_t)n * N_H / 4;
    int zgrid = (int)((agg4 + 255) / 256); if (zgrid > ZB) zgrid = ZB;
    zero_k<<<zgrid, 256, 0, stream>>>(agg, agg4);
    spmm<false><<<(e + 7) / 8, 256, 0, stream>>>(h, src, dst, perm, norm_out, agg, e);
    finalize_k<<<(n + 7) / 8, 256, 0, stream>>>(agg, norm_in, b, alpha, wsum, out, n);

    // 5) encoder 2 (corrupted): rows of h gathered through perm ; logit[n:2n]
    zero_k<<<zgrid, 256, 0, stream>>>(agg, agg4);
    spmm<true><<<(e + 7) / 8, 256, 0, stream>>>(h, src, dst, perm, norm_out, agg, e);
    finalize_k<<<(n + 7) / 8, 256, 0, stream>>>(agg, norm_in, b, alpha, wsum, out + n, n);
}